// SpatialTransformer_23029614641854
// MI455X (gfx1250) — compile-verified
//
#include <hip/hip_runtime.h>

// CDNA5 / gfx1250 spatial transformer (3D affine grid + trilinear sample).
// Coordinate transform done with V_WMMA_F32_16X16X4_F32 (f32 matrix core),
// interpolation is L2-resident float2 gathers.

typedef __attribute__((ext_vector_type(2))) float v2f;
typedef __attribute__((ext_vector_type(8))) float v8f;

#define OH 128
#define OW 128
#define OD 128
#define BATCH 4
#define VH 128
#define VW 128
#define VD 128

__global__ __launch_bounds__(256) void st3d_wmma_kernel(const float* __restrict__ vol,
                                                        const float* __restrict__ theta,
                                                        float* __restrict__ out) {
    const int j    = blockIdx.x;                 // OW index -> x_lin
    const int i    = blockIdx.y;                 // OH index -> y_lin
    const int pair = blockIdx.z;                 // batch pair {2*pair, 2*pair+1}
    const int tid  = threadIdx.x;
    const int wave = tid >> 5;                   // 8 waves per block
    const int lane = tid & 31;
    const int n    = lane & 15;                  // column within 16-wide tile
    const bool hi  = lane >= 16;
    const int k0   = wave * 16;                  // OD tile base
    const int b0   = pair * 2;

    const float step = 2.0f / 127.0f;            // jnp.linspace(-1,1,128) step

    // ---------------- A matrix (16x4 f32): rows 0-2 = theta[b0], rows 8-10 = theta[b1]
    // Layout: lane holds row M = lane%16; VGPR0 = K (lanes<16: 0, else 2), VGPR1 = K+1.
    const int kk = hi ? 2 : 0;
    float a0 = 0.0f, a1 = 0.0f;
    if (n < 3) {
        const float* tr = theta + (b0 * 3 + n) * 4;
        a0 = tr[kk];
        a1 = tr[kk + 1];
    } else if (n >= 8 && n < 11) {
        const float* tr = theta + ((b0 + 1) * 3 + (n - 8)) * 4;
        a0 = tr[kk];
        a1 = tr[kk + 1];
    }
    v2f Am;
    Am[0] = a0;
    Am[1] = a1;

    // ---------------- B matrix (4x16 f32): column n = (X, Y, Z, 1) for voxel (i,j,k0+n)
    // Mirrored striping: lane holds column N = lane%16; VGPR0 = K0/K2, VGPR1 = K1/K3.
    const float xg = -1.0f + (float)j * step;
    const float yg = -1.0f + (float)i * step;
    const float zg = -1.0f + (float)(k0 + n) * step;
    v2f Bm;
    Bm[0] = hi ? zg : xg;     // K=2 (Z) for high half, K=0 (X) for low half
    Bm[1] = hi ? 1.0f : yg;   // K=3 (1) for high half, K=1 (Y) for low half

    v8f Cm = {};
    // D = A x B + 0  -> rows 0..2 (lanes 0-15) = batch b0 coords,
    //                   rows 8..10 (lanes 16-31) = batch b1 coords.
    v8f Dm = __builtin_amdgcn_wmma_f32_16x16x4_f32(
        /*neg_a=*/false, Am, /*neg_b=*/false, Bm,
        /*c_mod=*/(short)0, Cm, /*reuse_a=*/false, /*reuse_b=*/false);

    float x = Dm[0];
    float y = Dm[1];
    float z = Dm[2];
    const int b = b0 + (hi ? 1 : 0);
    const int k = k0 + n;

    // ---------------- trilinear sampling, exact reference semantics ----------------
    x = 0.5f * (x + 1.0f) * (float)(VW - 1);
    y = 0.5f * (y + 1.0f) * (float)(VH - 1);
    z = 0.5f * (z + 1.0f) * (float)(VD - 1);

    int ix0 = (int)floorf(x), iy0 = (int)floorf(y), iz0 = (int)floorf(z);
    int ix1 = ix0 + 1, iy1 = iy0 + 1, iz1 = iz0 + 1;
    ix0 = min(max(ix0, 0), VW - 1); ix1 = min(max(ix1, 0), VW - 1);
    iy0 = min(max(iy0, 0), VH - 1); iy1 = min(max(iy1, 0), VH - 1);
    iz0 = min(max(iz0, 0), VD - 1); iz1 = min(max(iz1, 0), VD - 1);

    const float x0f = (float)ix0, x1f = (float)ix1;
    const float y0f = (float)iy0, y1f = (float)iy1;
    const float z0f = (float)iz0, z1f = (float)iz1;
    const float wx1 = x - x0f, wx0 = x1f - x;
    const float wy1 = y - y0f, wy0 = y1f - y;
    const float wz1 = z - z0f, wz0 = z1f - z;

    // vol shape (B,H,W,D,C=2): one float2 per voxel -> B64 gathers.
    const float2* __restrict__ v2 = reinterpret_cast<const float2*>(vol);
    const int base = b * (VH * VW * VD);
    const int py0  = base + iy0 * (VW * VD);
    const int py1  = base + iy1 * (VW * VD);
    const int px00 = py0 + ix0 * VD, px01 = py0 + ix1 * VD;
    const int px10 = py1 + ix0 * VD, px11 = py1 + ix1 * VD;

    const float2 g000 = v2[px00 + iz0];
    const float2 g001 = v2[px00 + iz1];
    const float2 g010 = v2[px01 + iz0];
    const float2 g011 = v2[px01 + iz1];
    const float2 g100 = v2[px10 + iz0];
    const float2 g101 = v2[px10 + iz1];
    const float2 g110 = v2[px11 + iz0];
    const float2 g111 = v2[px11 + iz1];

    const float w000 = wy0 * wx0 * wz0, w001 = wy0 * wx0 * wz1;
    const float w010 = wy0 * wx1 * wz0, w011 = wy0 * wx1 * wz1;
    const float w100 = wy1 * wx0 * wz0, w101 = wy1 * wx0 * wz1;
    const float w110 = wy1 * wx1 * wz0, w111 = wy1 * wx1 * wz1;

    float2 acc;
    acc.x = w000 * g000.x;
    acc.y = w000 * g000.y;
    acc.x = fmaf(w001, g001.x, acc.x); acc.y = fmaf(w001, g001.y, acc.y);
    acc.x = fmaf(w010, g010.x, acc.x); acc.y = fmaf(w010, g010.y, acc.y);
    acc.x = fmaf(w011, g011.x, acc.x); acc.y = fmaf(w011, g011.y, acc.y);
    acc.x = fmaf(w100, g100.x, acc.x); acc.y = fmaf(w100, g100.y, acc.y);
    acc.x = fmaf(w101, g101.x, acc.x); acc.y = fmaf(w101, g101.y, acc.y);
    acc.x = fmaf(w110, g110.x, acc.x); acc.y = fmaf(w110, g110.y, acc.y);
    acc.x = fmaf(w111, g111.x, acc.x); acc.y = fmaf(w111, g111.y, acc.y);

    // out shape (B, OH, OW, OD, 2): contiguous 128B stores per 16-lane half.
    float2* __restrict__ o2 = reinterpret_cast<float2*>(out);
    o2[((b * OH + i) * OW + j) * OD + k] = acc;
}

extern "C" void kernel_launch(void* const* d_in, const int* in_sizes, int n_in,
                              void* d_out, int out_size, void* d_ws, size_t ws_size,
                              hipStream_t stream) {
    const float* vol   = (const float*)d_in[0];   // (4,128,128,128,2) f32
    const float* theta = (const float*)d_in[1];   // (4,3,4) f32
    float* out         = (float*)d_out;           // (4,128,128,128,2) f32

    dim3 grid(OW, OH, BATCH / 2);                 // (128,128,2) blocks
    st3d_wmma_kernel<<<grid, 256, 0, stream>>>(vol, theta, out);
}